// Propagate_33208687133421
// MI455X (gfx1250) — compile-verified
//
#include <hip/hip_runtime.h>
#include <hip/hip_bf16.h>

// ---------------------------------------------------------------------------
// segment_sum( w[e] * x[src[e]] , dst[e] )  for  N=10000 nodes, E=640000,
// D=128 f32 features.  CSR-build + gather-accumulate (no float atomics).
// Hot gather uses gfx1250 async global->LDS data mover (depth-4 pipeline)
// synchronized with s_wait_asynccnt.
// ---------------------------------------------------------------------------

#define D_FEAT 128
#define WAVES_PER_BLOCK 8
#define PIPE 4  // async rows in flight per wave (power of two)

typedef int v4i __attribute__((ext_vector_type(4)));
typedef __attribute__((address_space(1))) v4i* g_v4i_p;  // ptr to global int4
typedef __attribute__((address_space(3))) v4i* l_v4i_p;  // ptr to LDS int4

// ---------------- CDNA5 async global->LDS helpers --------------------------
#if defined(__has_builtin)
#if __has_builtin(__builtin_amdgcn_global_load_async_to_lds_b128)
#define HAVE_ASYNC_B128 1
#endif
#if __has_builtin(__builtin_amdgcn_s_wait_asynccnt)
#define HAVE_WAIT_ASYNC 1
#endif
#endif

__device__ __forceinline__ void async_row_b128(const float* g, float* l) {
#if defined(HAVE_ASYNC_B128)
  __builtin_amdgcn_global_load_async_to_lds_b128(
      (g_v4i_p)g, (l_v4i_p)l, /*offset=*/0, /*cpol=*/0);
#else
  unsigned lds_addr =
      (unsigned)(unsigned long long)(__attribute__((address_space(3))) void*)l;
  unsigned long long gaddr = (unsigned long long)g;
  asm volatile("global_load_async_to_lds_b128 %0, %1, off"
               :
               : "v"(lds_addr), "v"(gaddr)
               : "memory");
#endif
}

template <int N>
__device__ __forceinline__ void wait_async_le() {
#if defined(HAVE_WAIT_ASYNC)
  __builtin_amdgcn_s_wait_asynccnt(N);
#else
  asm volatile("s_wait_asynccnt %0" ::"i"(N) : "memory");
#endif
}

// ---------------- phase 0: zero counts -------------------------------------
__global__ void zero_u_kernel(unsigned* p, int n) {
  int i = blockIdx.x * blockDim.x + threadIdx.x;
  if (i < n) p[i] = 0u;
}

// ---------------- phase 1: degree histogram of dst -------------------------
__global__ void count_kernel(const int* __restrict__ ei, unsigned* counts, int E) {
  int e = blockIdx.x * blockDim.x + threadIdx.x;
  if (e < E) atomicAdd(&counts[ei[E + e]], 1u);
}

// ---------------- phase 2: exclusive scan (single block) -------------------
__global__ __launch_bounds__(1024) void scan_kernel(const unsigned* __restrict__ counts,
                                                    unsigned* __restrict__ offsets,
                                                    unsigned* __restrict__ cursor, int n) {
  __shared__ unsigned sh[1024];
  const int t = threadIdx.x;
  const int IT = (n + 1024) >> 10;  // ceil((n+1)/1024)
  const int start = t * IT;

  unsigned local = 0;
  for (int i = 0; i < IT; ++i) {
    int idx = start + i;
    if (idx < n) local += counts[idx];
  }
  sh[t] = local;
  __syncthreads();
  for (int off = 1; off < 1024; off <<= 1) {
    unsigned v = 0;
    if (t >= off) v = sh[t - off];
    __syncthreads();
    sh[t] += v;
    __syncthreads();
  }
  unsigned running = (t == 0) ? 0u : sh[t - 1];
  for (int i = 0; i < IT; ++i) {
    int idx = start + i;
    if (idx <= n) {
      offsets[idx] = running;
      if (idx < n) {
        cursor[idx] = running;
        running += counts[idx];
      }
    }
  }
}

// ---------------- phase 3: bin edge ids by dst -----------------------------
__global__ void scatter_kernel(const int* __restrict__ ei, unsigned* cursor,
                               unsigned* __restrict__ eids, int E) {
  int e = blockIdx.x * blockDim.x + threadIdx.x;
  if (e < E) {
    int d = ei[E + e];
    unsigned pos = atomicAdd(&cursor[d], 1u);
    eids[pos] = (unsigned)e;
  }
}

// ---------------- phase 4: per-node gather-accumulate ----------------------
// One wave32 per node. Lane L owns feats [4L, 4L+4). Each edge's 512B x-row
// is DMA'd into a per-wave 4-deep LDS ring (async, 16B/lane), up to 4 rows in
// flight, then FMA'd from LDS. Metadata for the next 32-edge chunk is
// prefetched while the current chunk drains. Coalesced float4 stores; no
// float atomics.
__global__ __launch_bounds__(256) void gather_accum_kernel(
    const float* __restrict__ x, const int* __restrict__ ei,
    const float* __restrict__ w, const unsigned* __restrict__ offsets,
    const unsigned* __restrict__ eids, float* __restrict__ out, int n_nodes) {
  __shared__ float lbuf[WAVES_PER_BLOCK * PIPE * D_FEAT];  // 16 KB
  const int lane = threadIdx.x & 31;
  const int wave = threadIdx.x >> 5;
  const int node = blockIdx.x * WAVES_PER_BLOCK + wave;
  if (node >= n_nodes) return;

  const unsigned beg = offsets[node];
  const unsigned deg = offsets[node + 1] - beg;

  float ax = 0.f, ay = 0.f, az = 0.f, aw = 0.f;
  float* wbase = &lbuf[wave * (PIPE * D_FEAT)];

  // metadata for chunk 0: lane t holds (src, weight) of edge t
  int s_cur = 0;
  float w_cur = 0.f;
  {
    unsigned c0 = deg < 32u ? deg : 32u;
    if (lane < (int)c0) {
      unsigned e = eids[beg + lane];
      s_cur = ei[e];
      w_cur = w[e];
    }
  }

  const unsigned nchunks = (deg + 31u) >> 5;
  for (unsigned c = 0; c < nchunks; ++c) {
    const unsigned cb = c << 5;
    const unsigned rem = deg - cb;
    const unsigned cnt = rem < 32u ? rem : 32u;

    // prefetch next chunk's metadata early (overlaps with async pipeline)
    int s_nxt = 0;
    float w_nxt = 0.f;
    {
      unsigned nb = cb + 32u;
      if (nb < deg) {
        unsigned ncnt = deg - nb < 32u ? deg - nb : 32u;
        if (lane < (int)ncnt) {
          unsigned e = eids[beg + nb + lane];
          s_nxt = ei[e];
          w_nxt = w[e];
        }
      }
    }

    // prologue: fill the async ring with up to PIPE rows
    const unsigned pre = cnt < (unsigned)PIPE ? cnt : (unsigned)PIPE;
    for (unsigned k = 0; k < pre; ++k) {
      int sk = __shfl(s_cur, (int)k, 32);
      async_row_b128(x + (size_t)sk * D_FEAT + lane * 4,
                     wbase + (k & (PIPE - 1)) * D_FEAT + lane * 4);
    }

    for (unsigned k = 0; k < cnt; ++k) {
      if (k + PIPE < cnt) {  // keep PIPE rows in flight
        int sk = __shfl(s_cur, (int)(k + PIPE), 32);
        async_row_b128(x + (size_t)sk * D_FEAT + lane * 4,
                       wbase + ((k + PIPE) & (PIPE - 1)) * D_FEAT + lane * 4);
        wait_async_le<PIPE>();  // in-order completion => row k is in LDS
      } else {
        unsigned left = cnt - 1u - k;  // ops younger than k still outstanding
        if (left >= 3u)
          wait_async_le<3>();
        else if (left == 2u)
          wait_async_le<2>();
        else if (left == 1u)
          wait_async_le<1>();
        else
          wait_async_le<0>();
      }
      asm volatile("" ::: "memory");  // compiler barrier: re-read LDS
      const float4 v =
          *(const float4*)(wbase + (k & (PIPE - 1)) * D_FEAT + lane * 4);
      const float wk = __shfl(w_cur, (int)k, 32);
      ax = fmaf(wk, v.x, ax);
      ay = fmaf(wk, v.y, ay);
      az = fmaf(wk, v.z, az);
      aw = fmaf(wk, v.w, aw);
    }

    s_cur = s_nxt;
    w_cur = w_nxt;
  }

  float4 r;
  r.x = ax; r.y = ay; r.z = az; r.w = aw;
  *(float4*)(out + (size_t)node * D_FEAT + lane * 4) = r;
}

// ---------------- fallback path (workspace too small): plain atomics -------
__global__ void zero_f_kernel(float* p, int n) {
  int i = blockIdx.x * blockDim.x + threadIdx.x;
  if (i < n) p[i] = 0.f;
}

__global__ void atomic_scatter_kernel(const float* __restrict__ x,
                                      const int* __restrict__ ei,
                                      const float* __restrict__ w,
                                      float* out, int E) {
  int tid = blockIdx.x * blockDim.x + threadIdx.x;
  int e = tid >> 5;
  int lane = tid & 31;
  if (e < E) {
    int s = ei[e];
    int d = ei[E + e];
    float wv = w[e];
#pragma unroll
    for (int i = 0; i < 4; ++i) {
      int f = lane * 4 + i;
      atomicAdd(&out[(size_t)d * D_FEAT + f], wv * x[(size_t)s * D_FEAT + f]);
    }
  }
}

// ---------------------------------------------------------------------------
extern "C" void kernel_launch(void* const* d_in, const int* in_sizes, int n_in,
                              void* d_out, int out_size, void* d_ws, size_t ws_size,
                              hipStream_t stream) {
  const float* x = (const float*)d_in[0];
  const int* ei = (const int*)d_in[1];  // [2, E]: row 0 = src, row 1 = dst
  const float* w = (const float*)d_in[2];
  float* out = (float*)d_out;

  const int n_edges = in_sizes[2];
  const int n_nodes = in_sizes[0] / D_FEAT;

  const size_t need =
      ((size_t)n_nodes * 3 + 1 + (size_t)n_edges) * sizeof(unsigned);

  if (ws_size >= need) {
    unsigned* counts = (unsigned*)d_ws;          // [n_nodes]
    unsigned* offsets = counts + n_nodes;        // [n_nodes+1]
    unsigned* cursor = offsets + (n_nodes + 1);  // [n_nodes]
    unsigned* eids = cursor + n_nodes;           // [n_edges]

    zero_u_kernel<<<(n_nodes + 255) / 256, 256, 0, stream>>>(counts, n_nodes);
    count_kernel<<<(n_edges + 255) / 256, 256, 0, stream>>>(ei, counts, n_edges);
    scan_kernel<<<1, 1024, 0, stream>>>(counts, offsets, cursor, n_nodes);
    scatter_kernel<<<(n_edges + 255) / 256, 256, 0, stream>>>(ei, cursor, eids,
                                                              n_edges);
    gather_accum_kernel<<<(n_nodes + WAVES_PER_BLOCK - 1) / WAVES_PER_BLOCK, 256,
                          0, stream>>>(x, ei, w, offsets, eids, out, n_nodes);
  } else {
    int total = n_nodes * D_FEAT;
    zero_f_kernel<<<(total + 255) / 256, 256, 0, stream>>>(out, total);
    long long threads = (long long)n_edges * 32;
    atomic_scatter_kernel<<<(int)((threads + 255) / 256), 256, 0, stream>>>(
        x, ei, w, out, n_edges);
  }
}